// BoundaryPointChamferLoss_2714419331831
// MI455X (gfx1250) — compile-verified
//
#include <hip/hip_runtime.h>

typedef __attribute__((ext_vector_type(2))) float v2f;
typedef __attribute__((ext_vector_type(8))) float v8f;

#define N_PTS 4096
#define M_REF 32768

// ---------------------------------------------------------------------------
// Row pass: one wave32 per i-tile (16 pts rows). Streams all M/16 j-tiles.
// D = A x B via V_WMMA_F32_16X16X4_F32 where
//   A row m  = [ax, ay, ax^2+ay^2, 1]          (16x4, f32)
//   B col n  = [-2bx, -2by, 1, bx^2+by^2]      (4x16, f32)
// so D[m][n] = ||a_m - b_n||^2 directly.
// f32 A operand layout (ISA 7.12.2): lanes 0-15 hold K0/K1 in v0/v1,
// lanes 16-31 hold K2/K3. B mirrors with rows striped across lanes.
// ---------------------------------------------------------------------------
__global__ __launch_bounds__(256) void chamfer_row_kernel(
    const float* __restrict__ pts, const float* __restrict__ ref,
    float* __restrict__ row_out) {
  const int lane = threadIdx.x & 31;
  const int wave = (blockIdx.x * blockDim.x + threadIdx.x) >> 5; // i-tile id
  const int n    = lane & 15;
  const int hi   = lane >> 4;  // 0: lanes 0-15 (K0,K1) ; 1: lanes 16-31 (K2,K3)

  // A operand: loop-invariant. Lane handles point row i = wave*16 + n.
  const int i = wave * 16 + n;
  const float px = pts[2 * i + 0];
  const float py = pts[2 * i + 1];
  v2f a;
  a.x = hi ? (px * px + py * py) : px;  // K0 / K2
  a.y = hi ? 1.0f : py;                 // K1 / K3

  v8f rowacc;
#pragma unroll
  for (int r = 0; r < 8; ++r) rowacc[r] = __builtin_inff();

  for (int jt = 0; jt < M_REF / 16; ++jt) {
    const int j = jt * 16 + n;
    const float bx = ref[2 * j + 0];
    const float by = ref[2 * j + 1];
    v2f b;
    b.x = hi ? 1.0f : (-2.0f * bx);               // B row K0 / K2, col n
    b.y = hi ? (bx * bx + by * by) : (-2.0f * by); // B row K1 / K3, col n
    v8f c = {};
    v8f d = __builtin_amdgcn_wmma_f32_16x16x4_f32(
        /*neg_a=*/false, a, /*neg_b=*/false, b,
        /*c_mod=*/(short)0, c, /*reuse_a=*/false, /*reuse_b=*/false);
#pragma unroll
    for (int r = 0; r < 8; ++r) rowacc[r] = fminf(rowacc[r], d[r]);
  }

  // Min across the 16 lanes of each half-wave (row m fully reduced over j).
#pragma unroll
  for (int r = 0; r < 8; ++r) {
    float v = rowacc[r];
    v = fminf(v, __shfl_xor(v, 1, 32));
    v = fminf(v, __shfl_xor(v, 2, 32));
    v = fminf(v, __shfl_xor(v, 4, 32));
    v = fminf(v, __shfl_xor(v, 8, 32));
    rowacc[r] = v;
  }
  // VGPR r: lanes 0-15 -> row r, lanes 16-31 -> row r+8.
  if (n == 0) {
#pragma unroll
    for (int r = 0; r < 8; ++r) {
      const int row = wave * 16 + hi * 8 + r;
      row_out[row] = sqrtf(fmaxf(rowacc[r], 1e-12f));
    }
  }
}

// ---------------------------------------------------------------------------
// Col pass: one wave32 per j-tile (16 ref cols). Streams all N/16 i-tiles.
// Per tile, fold the 8 D-VGPRs into one scalar min per lane; lanes l / l+16
// cover complementary row halves, combined at the end with a lane-16 xor.
// ---------------------------------------------------------------------------
__global__ __launch_bounds__(256) void chamfer_col_kernel(
    const float* __restrict__ pts, const float* __restrict__ ref,
    float* __restrict__ col_out) {
  const int lane = threadIdx.x & 31;
  const int wave = (blockIdx.x * blockDim.x + threadIdx.x) >> 5; // j-tile id
  const int n    = lane & 15;
  const int hi   = lane >> 4;

  // B operand: loop-invariant. Lane handles ref column j = wave*16 + n.
  const int j = wave * 16 + n;
  const float bx = ref[2 * j + 0];
  const float by = ref[2 * j + 1];
  v2f b;
  b.x = hi ? 1.0f : (-2.0f * bx);
  b.y = hi ? (bx * bx + by * by) : (-2.0f * by);

  float colacc = __builtin_inff();

  for (int it = 0; it < N_PTS / 16; ++it) {
    const int i = it * 16 + n;
    const float px = pts[2 * i + 0];
    const float py = pts[2 * i + 1];
    v2f a;
    a.x = hi ? (px * px + py * py) : px;
    a.y = hi ? 1.0f : py;
    v8f c = {};
    v8f d = __builtin_amdgcn_wmma_f32_16x16x4_f32(
        false, a, false, b, (short)0, c, false, false);
#pragma unroll
    for (int r = 0; r < 8; ++r) colacc = fminf(colacc, d[r]);
  }

  // Lane l covers rows {16t+0..7}, lane l+16 covers {16t+8..15}: combine.
  colacc = fminf(colacc, __shfl_xor(colacc, 16, 32));
  if (lane < 16) col_out[wave * 16 + n] = sqrtf(fmaxf(colacc, 1e-12f));
}

// ---------------------------------------------------------------------------
// Deterministic final sum: one block, fixed strided order + LDS tree.
// ---------------------------------------------------------------------------
__global__ __launch_bounds__(256) void reduce_sum_kernel(
    const float* __restrict__ vals, int count, float* __restrict__ out) {
  __shared__ float smem[256];
  float s = 0.0f;
  for (int idx = threadIdx.x; idx < count; idx += 256) s += vals[idx];
  smem[threadIdx.x] = s;
  __syncthreads();
  for (int off = 128; off > 0; off >>= 1) {
    if ((int)threadIdx.x < off) smem[threadIdx.x] += smem[threadIdx.x + off];
    __syncthreads();
  }
  if (threadIdx.x == 0) out[0] = smem[0];
}

extern "C" void kernel_launch(void* const* d_in, const int* in_sizes, int n_in,
                              void* d_out, int out_size, void* d_ws, size_t ws_size,
                              hipStream_t stream) {
  const float* img  = (const float*)d_in[0];  // (2, 4096, 2): circle 0 at offset 0
  const float* skel = (const float*)d_in[1];  // (K, 32768, 2): last slice

  const float* pts = img;                                   // (4096, 2)
  const float* ref = skel + (in_sizes[1] - M_REF * 2);      // last slice, flip irrelevant

  float* ws      = (float*)d_ws;
  float* row_out = ws;            // N_PTS  sqrt'd row mins
  float* col_out = ws + N_PTS;    // M_REF  sqrt'd col mins

  // Row pass: 256 waves (one per 16-row tile) -> 32 blocks x 256 threads.
  chamfer_row_kernel<<<(N_PTS / 16) * 32 / 256, 256, 0, stream>>>(pts, ref, row_out);
  // Col pass: 2048 waves (one per 16-col tile) -> 256 blocks x 256 threads.
  chamfer_col_kernel<<<(M_REF / 16) * 32 / 256, 256, 0, stream>>>(pts, ref, col_out);
  // Final deterministic sum of N_PTS + M_REF values.
  reduce_sum_kernel<<<1, 256, 0, stream>>>(ws, N_PTS + M_REF, (float*)d_out);
}